// LengthRegulator_24790551232884
// MI455X (gfx1250) — compile-verified
//
#include <hip/hip_runtime.h>
#include <stdint.h>

// LengthRegulator expand: x[B,T,D] + durations[B,T] + gt[B] -> out[B,MAXF,D]
// B=32, T=256, D=512, MAXF=4096.  Store-bandwidth-bound (256 MB out, ~11us
// floor @ 23.3 TB/s).  Strategy: stage each distinct token row ONCE in LDS
// via async global->LDS loads, then fan it out to all its output frames with
// async LDS->global NT stores (no VGPR staging on the hot path).

#define NB      32
#define NTOK    256
#define ND      512
#define MAXF    4096
#define FPB     128        // frames per block
#define THREADS 256        // 8 wave32s
#define NWAVES  (THREADS / 32)
#define CAP     16         // cached token rows per block (2KB each) + 1 zero slot

typedef float v4f __attribute__((ext_vector_type(4)));

__device__ __forceinline__ int searchsorted_right(const float* cum, float mf) {
    int lo = 0, hi = NTOK;
    while (lo < hi) {
        const int mid = (lo + hi) >> 1;
        if (cum[mid] <= mf) lo = mid + 1; else hi = mid;
    }
    return lo;   // count(cum <= mf)
}

__global__ __launch_bounds__(THREADS) void lenreg_expand_kernel(
    const float* __restrict__ x,       // [B, T, D]
    const float* __restrict__ dur,     // [B, T]
    const int*   __restrict__ gt_num,  // [B]
    float*       __restrict__ out)     // [B, MAXF, D]
{
    __shared__ float s_a[NTOK];
    __shared__ float s_b[NTOK];
    __shared__ __align__(16) float s_cache[(CAP + 1) * ND];   // slot CAP = zeros

    const int b   = blockIdx.y;
    const int tid = threadIdx.x;

    // ---- 0) zero slot for invalid / past-gt frames ----
    if (tid < ND / 4) {
        ((v4f*)(s_cache + CAP * ND))[tid] = (v4f){0.f, 0.f, 0.f, 0.f};
    }

    // ---- 1) row sum of durations (tree reduction in LDS) ----
    const float v = dur[b * NTOK + tid];
    s_b[tid] = v;
    __syncthreads();
    for (int s = NTOK / 2; s > 0; s >>= 1) {
        if (tid < s) s_b[tid] += s_b[tid + s];
        __syncthreads();
    }
    const float total = s_b[0];
    const int   gti   = gt_num[b];
    const float gtf   = (float)gti;

    // ---- 2) normalize (reference op order: a/sum*gt), scan, clip ----
    s_a[tid] = v / total * gtf;
    __syncthreads();
    float* src = s_a;
    float* dst = s_b;
    for (int off = 1; off < NTOK; off <<= 1) {   // 8-step inclusive scan
        float val = src[tid];
        if (tid >= off) val += src[tid - off];
        dst[tid] = val;
        __syncthreads();
        float* t = src; src = dst; dst = t;
    }
    src[tid] = fminf(src[tid], gtf);             // clipped running cumsum
    __syncthreads();
    const float* cum = src;

    // ---- 3) token range covered by this block's frame chunk ----
    const int    m0       = blockIdx.x * FPB;
    const size_t x_row0   = (size_t)b * NTOK * ND;
    const size_t out_row0 = (size_t)b * MAXF * ND;
    const int    wave     = tid >> 5;
    const int    lane     = tid & 31;

    int t0 = 0, ncached = 0;
    if (m0 < gti) {
        t0 = searchsorted_right(cum, (float)m0);
        const int mh = min(m0 + FPB - 1, gti - 1);
        int t1 = searchsorted_right(cum, (float)mh);
        if (t1 > NTOK - 1) t1 = NTOK - 1;
        if (t0 <= NTOK - 1) {
            ncached = t1 - t0 + 1;
            if (ncached > CAP) ncached = CAP;
        }
    }

    // ---- 4) stage rows [t0, t0+ncached) into LDS: async global->LDS ----
    for (int i = wave; i < ncached; i += NWAVES) {
        const uint64_t gbase = (uint64_t)(uintptr_t)(x + x_row0 + (size_t)(t0 + i) * ND);
        const uint32_t lbase = (uint32_t)(uintptr_t)(s_cache + i * ND);
        #pragma unroll
        for (int k = 0; k < 4; ++k) {                 // 4 x (32 lanes x 16B) = 2KB
            const uint64_t ga = gbase + (uint64_t)(lane * 16 + k * 512);
            const uint32_t la = lbase + (uint32_t)(lane * 16 + k * 512);
            asm volatile("global_load_async_to_lds_b128 %0, %1, off"
                         :: "v"(la), "v"(ga) : "memory");
        }
    }
    asm volatile("s_wait_asynccnt 0x0" ::: "memory");  // my loads landed in LDS
    __syncthreads();                                   // everyone's loads landed

    // ---- 5) expand: one wave per frame; fan out rows LDS -> global ----
    for (int f = wave; f < FPB; f += NWAVES) {
        const int    m    = m0 + f;
        const float* orow = out + out_row0 + (size_t)m * ND;

        int slot;                 // LDS slot to stream out, or -1 => fallback
        int idx = 0;
        if (m < gti) {
            idx = searchsorted_right(cum, (float)m);   // uniform per wave
            if (idx >= NTOK)               slot = CAP;       // clipped tail -> zeros
            else if (idx - t0 < ncached)   slot = idx - t0;  // cached row
            else                           slot = -1;        // beyond cache
        } else {
            slot = CAP;                                      // past gt -> zeros
        }

        if (slot >= 0) {
            // async LDS->global streaming store, no VGPR data path
            const uint64_t gbase = (uint64_t)(uintptr_t)orow;
            const uint32_t lbase = (uint32_t)(uintptr_t)(s_cache + slot * ND);
            #pragma unroll
            for (int k = 0; k < 4; ++k) {
                const uint64_t ga = gbase + (uint64_t)(lane * 16 + k * 512);
                const uint32_t la = lbase + (uint32_t)(lane * 16 + k * 512);
                asm volatile("global_store_async_from_lds_b128 %0, %1, off th:TH_STORE_NT"
                             :: "v"(ga), "v"(la) : "memory");
            }
        } else {
            // rare fallback: VGPR gather + non-temporal b128 stores
            const v4f* s4 = (const v4f*)(x + x_row0 + (size_t)idx * ND);
            const v4f r0 = s4[lane];
            const v4f r1 = s4[lane + 32];
            const v4f r2 = s4[lane + 64];
            const v4f r3 = s4[lane + 96];
            v4f* d4 = (v4f*)orow;
            __builtin_nontemporal_store(r0, d4 + lane);
            __builtin_nontemporal_store(r1, d4 + lane + 32);
            __builtin_nontemporal_store(r2, d4 + lane + 64);
            __builtin_nontemporal_store(r3, d4 + lane + 96);
        }
    }
    // S_ENDPGM performs an implicit wait-idle, draining outstanding async
    // stores before the wave (and its LDS) is released.
}

extern "C" void kernel_launch(void* const* d_in, const int* in_sizes, int n_in,
                              void* d_out, int out_size, void* d_ws, size_t ws_size,
                              hipStream_t stream) {
    (void)in_sizes; (void)n_in; (void)out_size; (void)d_ws; (void)ws_size;
    const float* x   = (const float*)d_in[0];   // [B,T,D] f32
    const float* dur = (const float*)d_in[1];   // [B,T]   f32
    const int*   gt  = (const int*)  d_in[2];   // [B]     i32
    // d_in[3] (token_nums) is unused by the reference forward.
    float* out = (float*)d_out;                 // [B,MAXF,D] f32

    dim3 grid(MAXF / FPB, NB);                  // 32 x 32 = 1024 blocks
    lenreg_expand_kernel<<<grid, THREADS, 0, stream>>>(x, dur, gt, out);
}